// ExpertChoiceRouter_67284957659235
// MI455X (gfx1250) — compile-verified
//
#include <hip/hip_runtime.h>
#include <hip/hip_bf16.h>
#include <math.h>

// ---------------------------------------------------------------------------
// ExpertChoiceRouter for MI455X (gfx1250, wave32)
//   logits = x @ W^T + b            (16384 x 64)   -- V_WMMA_F32_16X16X4_F32
//   probs  = softmax over tokens    (column softmax)
//   top-512 per expert (sorted desc, with indices) -- 128KB-LDS bitonic sort
// ---------------------------------------------------------------------------

#define NUM_TOKENS 16384
#define D_MODEL    2048
#define NUM_EXPERTS 64
#define CAPACITY   512

typedef __attribute__((ext_vector_type(2))) float v2f;
typedef __attribute__((ext_vector_type(8))) float v8f;

// ---------------------------------------------------------------------------
// Kernel 1: router GEMM via f32 WMMA.
// Block = 256 threads = 8 waves. Each wave owns a 16(token) x 64(expert)
// strip: 4 accumulator tiles of 16x16. K loop steps by 4 (WMMA K=4).
//
// A (16x4 f32) per-lane layout (ISA 7.12.2):
//   lanes 0-15: M=lane,   VGPR0=K0, VGPR1=K1
//   lanes16-31: M=lane-16,VGPR0=K2, VGPR1=K3
// => per lane one float2 load at x[(row0+m)*D + k0 + half*2]
// B (4x16 f32) mirrored over N:
//   lane holds column N=lane&15, K pair selected by lane>>4
// => per lane one float2 load at W[(n0+m)*D + k0 + half*2]   (W is E x D)
// C/D (16x16 f32): VGPR r, lane l -> M = r + (l>>4)*8, N = l&15.
// ---------------------------------------------------------------------------
__global__ void __launch_bounds__(256)
router_gemm_wmma(const float* __restrict__ x, const float* __restrict__ W,
                 const float* __restrict__ bias, float* __restrict__ logits) {
  const int lane = threadIdx.x & 31;
  const int wave = threadIdx.x >> 5;
  const int m    = lane & 15;       // M (for A) / N (for B) sub-index
  const int half = lane >> 4;       // selects K pair {0,1} vs {2,3}
  const int row0 = blockIdx.x * 128 + wave * 16;

  const float* __restrict__ aRow = x + (size_t)(row0 + m) * D_MODEL + half * 2;
  // 4 expert tiles: n-tile t covers experts [t*16, t*16+16)
  const float* __restrict__ bRow0 = W + (size_t)(0 * 16 + m) * D_MODEL + half * 2;
  const float* __restrict__ bRow1 = W + (size_t)(1 * 16 + m) * D_MODEL + half * 2;
  const float* __restrict__ bRow2 = W + (size_t)(2 * 16 + m) * D_MODEL + half * 2;
  const float* __restrict__ bRow3 = W + (size_t)(3 * 16 + m) * D_MODEL + half * 2;

  v8f acc0 = {}, acc1 = {}, acc2 = {}, acc3 = {};

#pragma unroll 4
  for (int k0 = 0; k0 < D_MODEL; k0 += 4) {
    v2f a  = *(const v2f*)(aRow  + k0);
    v2f b0 = *(const v2f*)(bRow0 + k0);
    v2f b1 = *(const v2f*)(bRow1 + k0);
    v2f b2 = *(const v2f*)(bRow2 + k0);
    v2f b3 = *(const v2f*)(bRow3 + k0);
    acc0 = __builtin_amdgcn_wmma_f32_16x16x4_f32(false, a, false, b0,
                                                 (short)0, acc0, false, false);
    acc1 = __builtin_amdgcn_wmma_f32_16x16x4_f32(false, a, false, b1,
                                                 (short)0, acc1, false, false);
    acc2 = __builtin_amdgcn_wmma_f32_16x16x4_f32(false, a, false, b2,
                                                 (short)0, acc2, false, false);
    acc3 = __builtin_amdgcn_wmma_f32_16x16x4_f32(false, a, false, b3,
                                                 (short)0, acc3, false, false);
  }

  // Epilogue: add bias, store logits. Lane l, VGPR r -> M=r+half*8, N=l&15.
  const int n = lane & 15;
  float bv0 = bias[0 * 16 + n];
  float bv1 = bias[1 * 16 + n];
  float bv2 = bias[2 * 16 + n];
  float bv3 = bias[3 * 16 + n];
#pragma unroll
  for (int r = 0; r < 8; ++r) {
    const int row = row0 + r + half * 8;
    float* o = logits + (size_t)row * NUM_EXPERTS;
    o[0 * 16 + n] = acc0[r] + bv0;
    o[1 * 16 + n] = acc1[r] + bv1;
    o[2 * 16 + n] = acc2[r] + bv2;
    o[3 * 16 + n] = acc3[r] + bv3;
  }
}

// ---------------------------------------------------------------------------
// Kernel 2: per-expert column max and sum(exp(l - max)) over 16384 tokens.
// One block per expert; logits are L2-resident (4 MB << 192 MB L2).
// ---------------------------------------------------------------------------
__global__ void __launch_bounds__(256)
router_colstats(const float* __restrict__ logits, float* __restrict__ colmax,
                float* __restrict__ colsum) {
  __shared__ float red[256];
  const int e = blockIdx.x;
  const int tid = threadIdx.x;

  float m = -INFINITY;
  for (int t = tid; t < NUM_TOKENS; t += 256)
    m = fmaxf(m, logits[(size_t)t * NUM_EXPERTS + e]);
  red[tid] = m;
  __syncthreads();
  for (int s = 128; s > 0; s >>= 1) {
    if (tid < s) red[tid] = fmaxf(red[tid], red[tid + s]);
    __syncthreads();
  }
  const float mx = red[0];
  __syncthreads();

  float sum = 0.f;
  for (int t = tid; t < NUM_TOKENS; t += 256)
    sum += expf(logits[(size_t)t * NUM_EXPERTS + e] - mx);
  red[tid] = sum;
  __syncthreads();
  for (int s = 128; s > 0; s >>= 1) {
    if (tid < s) red[tid] += red[tid + s];
    __syncthreads();
  }
  if (tid == 0) {
    colmax[e] = mx;
    colsum[e] = red[0];
  }
}

// ---------------------------------------------------------------------------
// Kernel 3: probs[t,e] = exp(logits[t,e] - colmax[e]) / colsum[e]
// ---------------------------------------------------------------------------
__global__ void __launch_bounds__(256)
router_probs(const float* __restrict__ logits, const float* __restrict__ colmax,
             const float* __restrict__ colsum, float* __restrict__ probs) {
  const int idx = blockIdx.x * 256 + threadIdx.x;
  const int e = idx & (NUM_EXPERTS - 1);
  probs[idx] = expf(logits[idx] - colmax[e]) / colsum[e];
}

// ---------------------------------------------------------------------------
// Kernel 4: per-expert top-512 of 16384, sorted descending, jax tie-break
// (lower index wins on equal prob). probs are >= 0 so their IEEE bits are
// order-preserving as u32. key = (prob_bits << 32) | ~index  => plain u64
// descending sort gives exactly jax.lax.top_k ordering.
// Full 16384-element bitonic sort in 128 KB of dynamic LDS (CDNA5: 320 KB/WGP).
// ---------------------------------------------------------------------------
__global__ void __launch_bounds__(256)
router_topk(const float* __restrict__ probs, float* __restrict__ expert_probs,
            float* __restrict__ expert_indices) {
  extern __shared__ unsigned long long keys[];   // 16384 * 8B = 128 KB
  const int e = blockIdx.x;
  const int tid = threadIdx.x;

  for (int i = tid; i < NUM_TOKENS; i += 256) {
    unsigned bits = __float_as_uint(probs[(size_t)i * NUM_EXPERTS + e]);
    keys[i] = ((unsigned long long)bits << 32) | (unsigned)(~(unsigned)i);
  }

  for (unsigned k = 2; k <= NUM_TOKENS; k <<= 1) {
    for (unsigned j = k >> 1; j > 0; j >>= 1) {
      __syncthreads();
      for (unsigned i = tid; i < NUM_TOKENS; i += 256) {
        const unsigned partner = i ^ j;
        if (partner > i) {
          const bool desc = ((i & k) == 0);   // final merge (k==N) => all desc
          unsigned long long a = keys[i];
          unsigned long long p = keys[partner];
          const bool do_swap = desc ? (a < p) : (a > p);
          if (do_swap) {
            keys[i] = p;
            keys[partner] = a;
          }
        }
      }
    }
  }
  __syncthreads();

  for (int r = tid; r < CAPACITY; r += 256) {
    const unsigned long long key = keys[r];
    const float p = __uint_as_float((unsigned)(key >> 32));
    const unsigned idx = ~(unsigned)(key & 0xFFFFFFFFull);
    expert_probs[(size_t)e * CAPACITY + r] = p;
    expert_indices[(size_t)e * CAPACITY + r] = (float)idx;
  }
}

// ---------------------------------------------------------------------------
extern "C" void kernel_launch(void* const* d_in, const int* in_sizes, int n_in,
                              void* d_out, int out_size, void* d_ws, size_t ws_size,
                              hipStream_t stream) {
  (void)in_sizes; (void)n_in; (void)out_size; (void)ws_size;
  const float* x = (const float*)d_in[0];
  const float* W = (const float*)d_in[1];
  const float* b = (const float*)d_in[2];

  float* out = (float*)d_out;
  float* logits = out;                                           // 16384*64
  float* probs  = out + (size_t)NUM_TOKENS * NUM_EXPERTS;        // 16384*64
  float* eprobs = probs + (size_t)NUM_TOKENS * NUM_EXPERTS;      // 64*512
  float* eidx   = eprobs + (size_t)NUM_EXPERTS * CAPACITY;       // 64*512

  float* colmax = (float*)d_ws;              // 64 floats
  float* colsum = colmax + NUM_EXPERTS;      // 64 floats

  // 1) GEMM: 128 tokens per block (8 waves x 16 rows), 128 blocks.
  router_gemm_wmma<<<NUM_TOKENS / 128, 256, 0, stream>>>(x, W, b, logits);

  // 2) column softmax statistics: one block per expert.
  router_colstats<<<NUM_EXPERTS, 256, 0, stream>>>(logits, colmax, colsum);

  // 3) probs, elementwise over 1M entries.
  router_probs<<<(NUM_TOKENS * NUM_EXPERTS) / 256, 256, 0, stream>>>(
      logits, colmax, colsum, probs);

  // 4) top-512 per expert: one block per expert, 128 KB dynamic LDS.
  router_topk<<<NUM_EXPERTS, 256, NUM_TOKENS * sizeof(unsigned long long),
                stream>>>(probs, eprobs, eidx);
}